// ChebEncoder_26061861552298
// MI455X (gfx1250) — compile-verified
//
#include <hip/hip_runtime.h>

// ---------------------------------------------------------------------------
// ChebNet encoder for MI455X (gfx1250, wave32).
//  - GEMMs: V_WMMA_F32_16X16X4_F32 (fp32 tensor path; net is HBM-bound).
//  - Weight panel staged into LDS with GLOBAL_LOAD_ASYNC_TO_LDS_B128
//    (ASYNCcnt double-buffered pipeline), A streamed from HBM.
//  - SpMM: flat edge scatter with global_atomic_add_f32.
// ---------------------------------------------------------------------------

typedef float v2f __attribute__((ext_vector_type(2)));
typedef float v8f __attribute__((ext_vector_type(8)));

#define TPB 256
static inline unsigned cdiv_u(size_t a, unsigned b) { return (unsigned)((a + b - 1) / b); }

// ---------------- elementwise helpers ----------------

__global__ void k_scalecopy(const float* __restrict__ src, int lds,
                            float* __restrict__ dst, int ldd,
                            int C, float scale, size_t total) {
    size_t t = (size_t)blockIdx.x * blockDim.x + threadIdx.x;
    if (t >= total) return;
    size_t m = t / C; int i = (int)(t % C);
    dst[m * (size_t)ldd + i] = scale * src[m * (size_t)lds + i];
}

__global__ void k_zero_slot(float* __restrict__ dst, int ldd, int C, size_t total) {
    size_t t = (size_t)blockIdx.x * blockDim.x + threadIdx.x;
    if (t >= total) return;
    size_t m = t / C; int i = (int)(t % C);
    dst[m * (size_t)ldd + i] = 0.0f;
}

// x (B, C, V) -> Xcat slot0 (row m=v*B+b, col i), row stride ld
__global__ void k_gather_input(const float* __restrict__ x, float* __restrict__ Xc,
                               int V, int B, int C, int ld, size_t total) {
    size_t t = (size_t)blockIdx.x * blockDim.x + threadIdx.x;
    if (t >= total) return;
    int i = (int)(t % C); size_t r = t / C;
    int b = (int)(r % B); size_t v = r / B;
    Xc[(v * B + b) * (size_t)ld + i] = x[((size_t)b * C + i) * V + v];
}

// scatter-add:  dst[(rows[e]*B+b)*ld + i] += scale*vals[e]*src[(cols[e]*B+b)*ld + i]
__global__ void k_spmm_edges(const int* __restrict__ rows, const int* __restrict__ cols,
                             const float* __restrict__ vals, int E,
                             int B, int C, int ld,
                             const float* __restrict__ src, float* __restrict__ dst,
                             float scale, size_t total) {
    size_t t = (size_t)blockIdx.x * blockDim.x + threadIdx.x;
    if (t >= total) return;
    int BC = B * C;
    int e = (int)(t / BC); int f = (int)(t % BC);
    int b = f / C; int i = f % C;
    float w = scale * vals[e];
    float z = src[((size_t)cols[e] * B + b) * (size_t)ld + i];
    atomicAdd(&dst[((size_t)rows[e] * B + b) * (size_t)ld + i], w * z);
}

// avg-pool pairs of vertices: out (V/2,B,C) row-major <- in (V,B,C)
__global__ void k_pool2(const float* __restrict__ in, float* __restrict__ out,
                        int Vh, int B, int C, size_t total) {
    size_t t = (size_t)blockIdx.x * blockDim.x + threadIdx.x;
    if (t >= total) return;
    int c = (int)(t % C); size_t r = t / C;
    int b = (int)(r % B); size_t v = r / B;
    size_t i0 = ((2 * v) * (size_t)B + b) * C + c;
    size_t i1 = ((2 * v + 1) * (size_t)B + b) * C + c;
    out[t] = 0.5f * (in[i0] + in[i1]);
}

// h (V,B,C) row-major -> out (B,C,V)
__global__ void k_store_out(const float* __restrict__ h, float* __restrict__ out,
                            int V, int B, int C, size_t total) {
    size_t t = (size_t)blockIdx.x * blockDim.x + threadIdx.x;
    if (t >= total) return;
    int c = (int)(t % C); size_t r = t / C;
    int b = (int)(r % B); size_t v = r / B;
    out[((size_t)b * C + c) * V + v] = h[t];
}

// ---------------- async global->LDS staging (gfx1250) ----------------

// One 16-byte lane copy: global -> LDS, tracked by ASYNCcnt (no VGPR data).
__device__ __forceinline__ void async_copy_b128(const float* g, unsigned lds_byte_off) {
    asm volatile("global_load_async_to_lds_b128 %0, %1, off"
                 :: "v"(lds_byte_off), "v"(g) : "memory");
}
__device__ __forceinline__ void wait_async0() {
    asm volatile("s_wait_asynccnt 0x0" ::: "memory");
}

// Stage kb rows x NTILE cols of B(row-major, ldb=N) into lds (row stride NTILE).
template <int NTILE>
__device__ __forceinline__ void stage_B(const float* __restrict__ Bw, int N, int n0,
                                        int k0, int kb, float* lds) {
    const int nq  = NTILE / 4;      // 16B chunks per row
    const int cnt = kb * nq;
    for (int idx = threadIdx.x; idx < cnt; idx += TPB) {
        int row  = idx / nq;
        int col4 = (idx - row * nq) * 4;
        const float* g = Bw + (size_t)(k0 + row) * N + n0 + col4;
        unsigned loff = (unsigned)(uintptr_t)(lds + row * NTILE + col4);
        async_copy_b128(g, loff);
    }
}

// ---------------- WMMA fp32 GEMM ----------------
// C(MxN) = A(MxKC, lda=KC) * B(KCxN, ldb=N)  [+ bias, epilogue EP]
// EP=0: C = acc + bias ; EP=1: C = relu(acc+bias) ; EP=2: C = relu(acc+bias+C)
// Block = 256 threads (8 wave32). Wave w owns M-tile (blockIdx.x*8+w)*16;
// NT accumulators cover the block's N-tile of NT*16 at blockIdx.y*NT*16.
// B panel is double-buffered through LDS in KB-row chunks via async copies.
#define KB 32
template <int NT, int EP>
__global__ void k_gemm_wmma_f32(const float* __restrict__ A, const float* __restrict__ Bw,
                                const float* __restrict__ bias, float* __restrict__ C,
                                int M, int N, int KC) {
    constexpr int NTILE = NT * 16;
    __shared__ float ldsB[2][KB * 64];

    const int lane  = threadIdx.x & 31;
    const int wave  = threadIdx.x >> 5;
    const int lrow  = lane & 15;   // M-row (A) / N-col (B,C) within 16-tile
    const int lhalf = lane >> 4;   // selects K pair (A/B) and C row half
    const int m0 = (blockIdx.x * 8 + wave) * 16;
    const int n0 = blockIdx.y * NTILE;

    v8f acc[NT];
#pragma unroll
    for (int s = 0; s < NT; ++s)
#pragma unroll
        for (int r = 0; r < 8; ++r) acc[s][r] = 0.0f;

    // prologue: stage chunk 0
    stage_B<NTILE>(Bw, N, n0, 0, (KC < KB ? KC : KB), ldsB[0]);

    const float* ap = A + (size_t)(m0 + lrow) * KC + 2 * lhalf;
    const int nch = (KC + KB - 1) / KB;
    for (int c = 0; c < nch; ++c) {
        wait_async0();       // own async copies into ldsB[c&1] complete
        __syncthreads();     // all waves' copies visible; prev buffer free

        const int k0n = (c + 1) * KB;
        if (k0n < KC) {
            int kbn = KC - k0n; if (kbn > KB) kbn = KB;
            stage_B<NTILE>(Bw, N, n0, k0n, kbn, ldsB[(c + 1) & 1]);
        }

        const float* lb = ldsB[c & 1];
        int kb = KC - c * KB; if (kb > KB) kb = KB;
        const float* apc = ap + c * KB;
        for (int kk = 0; kk < kb; kk += 4) {
            v2f a;
            a.x = apc[kk];
            a.y = apc[kk + 1];
#pragma unroll
            for (int s = 0; s < NT; ++s) {
                const float* bp = lb + (kk + 2 * lhalf) * NTILE + s * 16 + lrow;
                v2f b;
                b.x = bp[0];
                b.y = bp[NTILE];
                acc[s] = __builtin_amdgcn_wmma_f32_16x16x4_f32(
                    false, a, false, b, (short)0, acc[s], false, false);
            }
        }
    }

    // C layout: lane<16 -> N=lrow, M rows 0..7 in acc[.][r]; lane>=16 -> rows 8..15
#pragma unroll
    for (int s = 0; s < NT; ++s) {
        const int n = n0 + s * 16 + lrow;
        const float bv = bias[n];
#pragma unroll
        for (int r = 0; r < 8; ++r) {
            size_t idx = (size_t)(m0 + lhalf * 8 + r) * N + n;
            float v = acc[s][r] + bv;
            if (EP == 1) v = fmaxf(v, 0.0f);
            if (EP == 2) v = fmaxf(v + C[idx], 0.0f);
            C[idx] = v;
        }
    }
}

static void gemm_launch(int ep, const float* A, const float* Bw, const float* bias,
                        float* C, int M, int N, int KC, hipStream_t s) {
    dim3 blk(256);
    if (N >= 64) {
        dim3 g(M / 128, N / 64);
        if (ep == 0)      k_gemm_wmma_f32<4, 0><<<g, blk, 0, s>>>(A, Bw, bias, C, M, N, KC);
        else if (ep == 1) k_gemm_wmma_f32<4, 1><<<g, blk, 0, s>>>(A, Bw, bias, C, M, N, KC);
        else              k_gemm_wmma_f32<4, 2><<<g, blk, 0, s>>>(A, Bw, bias, C, M, N, KC);
    } else if (N == 32) {
        dim3 g(M / 128, 1);
        if (ep == 0)      k_gemm_wmma_f32<2, 0><<<g, blk, 0, s>>>(A, Bw, bias, C, M, N, KC);
        else if (ep == 1) k_gemm_wmma_f32<2, 1><<<g, blk, 0, s>>>(A, Bw, bias, C, M, N, KC);
        else              k_gemm_wmma_f32<2, 2><<<g, blk, 0, s>>>(A, Bw, bias, C, M, N, KC);
    } else { // N == 16
        dim3 g(M / 128, 1);
        if (ep == 0)      k_gemm_wmma_f32<1, 0><<<g, blk, 0, s>>>(A, Bw, bias, C, M, N, KC);
        else if (ep == 1) k_gemm_wmma_f32<1, 1><<<g, blk, 0, s>>>(A, Bw, bias, C, M, N, KC);
        else              k_gemm_wmma_f32<1, 2><<<g, blk, 0, s>>>(A, Bw, bias, C, M, N, KC);
    }
}

// ---------------------------------------------------------------------------

extern "C" void kernel_launch(void* const* d_in, const int* in_sizes, int n_in,
                              void* d_out, int out_size, void* d_ws, size_t ws_size,
                              hipStream_t stream) {
    (void)in_sizes; (void)n_in; (void)out_size; (void)ws_size;
    const int B = 8;

    // workspace layout (floats); requires ~218 MB
    float* ws_f = (float*)d_ws;
    float* XCAT = ws_f;                          // up to 33,554,432 floats (M x 4*Cmax)
    float* H1   = XCAT + (size_t)33554432;       // up to  8,388,608 floats (M x Co)
    float* HOUT = H1   + (size_t)8388608;        // up to  8,388,608 floats (M x Co)
    float* XIN  = HOUT + (size_t)8388608;        // up to  4,194,304 floats (M x Ci)
    float* outp = (float*)d_out;

    const float* x = (const float*)d_in[0];

    const int*   rowsL[6]; const int* colsL[6]; const float* valsL[6];
    for (int j = 0; j < 6; ++j) {
        rowsL[j] = (const int*)d_in[1 + 3 * j];
        colsL[j] = (const int*)d_in[2 + 3 * j];
        valsL[j] = (const float*)d_in[3 + 3 * j];
    }
    const float* w_in = (const float*)d_in[19];
    const float* b_in = (const float*)d_in[20];

    const int wi1[6] = {21, 27, 33, 39, 45, 49}, bi1[6] = {22, 28, 34, 40, 46, 50};
    const int wi2[6] = {23, 29, 35, 41, 47, 51}, bi2[6] = {24, 30, 36, 42, 48, 52};
    const int wis[6] = {25, 31, 37, 43, -1, -1}, bis[6] = {26, 32, 38, 44, -1, -1};

    const int Vl[6]  = {32768, 16384, 8192, 4096, 2048, 1024};
    const int CiL[6] = {16, 32, 64, 128, 256, 256};
    const int CoL[6] = {32, 64, 128, 256, 256, 256};
    const size_t oofs[6] = {31457280, 23068672, 14680064, 6291456, 2097152, 0};

    auto cheb_build = [&](float* Xc, int M, int C, int ld, int j) {
        const int E = Vl[j] * 8;
        size_t totMC = (size_t)M * C;
        size_t totE  = (size_t)E * B * C;
        // T1 = L*T0
        k_zero_slot<<<cdiv_u(totMC, TPB), TPB, 0, stream>>>(Xc + C, ld, C, totMC);
        k_spmm_edges<<<cdiv_u(totE, TPB), TPB, 0, stream>>>(rowsL[j], colsL[j], valsL[j], E,
                                                            B, C, ld, Xc, Xc + C, 1.0f, totE);
        // T2 = 2*L*T1 - T0
        k_scalecopy<<<cdiv_u(totMC, TPB), TPB, 0, stream>>>(Xc, ld, Xc + 2 * C, ld, C, -1.0f, totMC);
        k_spmm_edges<<<cdiv_u(totE, TPB), TPB, 0, stream>>>(rowsL[j], colsL[j], valsL[j], E,
                                                            B, C, ld, Xc + C, Xc + 2 * C, 2.0f, totE);
        // T3 = 2*L*T2 - T1
        k_scalecopy<<<cdiv_u(totMC, TPB), TPB, 0, stream>>>(Xc + C, ld, Xc + 3 * C, ld, C, -1.0f, totMC);
        k_spmm_edges<<<cdiv_u(totE, TPB), TPB, 0, stream>>>(rowsL[j], colsL[j], valsL[j], E,
                                                            B, C, ld, Xc + 2 * C, Xc + 3 * C, 2.0f, totE);
    };

    // ---- input cheb: h = relu(cheb(xt, w_in, b_in)) -> XIN (M5 x 16) ----
    {
        const int V = Vl[0], C = 3, ld = 12, M = V * B;
        size_t tot = (size_t)M * C;
        k_gather_input<<<cdiv_u(tot, TPB), TPB, 0, stream>>>(x, XCAT, V, B, C, ld, tot);
        cheb_build(XCAT, M, C, ld, 0);
        gemm_launch(1, XCAT, w_in, b_in, XIN, M, 16, 12, stream);
    }

    // ---- residual blocks, level 5 down to 0 ----
    for (int j = 0; j < 6; ++j) {
        const int V = Vl[j], Ci = CiL[j], Co = CoL[j], M = V * B;
        const float* w1 = (const float*)d_in[wi1[j]];
        const float* b1 = (const float*)d_in[bi1[j]];
        const float* w2 = (const float*)d_in[wi2[j]];
        const float* b2 = (const float*)d_in[bi2[j]];

        // cheb1: H1 = relu(cheb(XIN) @ w1 + b1)
        {
            const int ld = 4 * Ci;
            size_t tot = (size_t)M * Ci;
            k_scalecopy<<<cdiv_u(tot, TPB), TPB, 0, stream>>>(XIN, Ci, XCAT, ld, Ci, 1.0f, tot);
            cheb_build(XCAT, M, Ci, ld, j);
            gemm_launch(1, XCAT, w1, b1, H1, M, Co, 4 * Ci, stream);
        }

        // skip path into HOUT
        if (wis[j] >= 0) {
            gemm_launch(0, XIN, (const float*)d_in[wis[j]], (const float*)d_in[bis[j]],
                        HOUT, M, Co, Ci, stream);
        } else { // identity skip (Ci == Co)
            size_t tot = (size_t)M * Co;
            k_scalecopy<<<cdiv_u(tot, TPB), TPB, 0, stream>>>(XIN, Ci, HOUT, Co, Co, 1.0f, tot);
        }

        // cheb2 fused with residual+relu: HOUT = relu(cheb(H1) @ w2 + b2 + HOUT)
        {
            const int ld = 4 * Co;
            size_t tot = (size_t)M * Co;
            k_scalecopy<<<cdiv_u(tot, TPB), TPB, 0, stream>>>(H1, Co, XCAT, ld, Co, 1.0f, tot);
            cheb_build(XCAT, M, Co, ld, j);
            gemm_launch(2, XCAT, w2, b2, HOUT, M, Co, 4 * Co, stream);
        }

        // emit e_l as (B, Co, V)
        {
            size_t tot = (size_t)V * B * Co;
            k_store_out<<<cdiv_u(tot, TPB), TPB, 0, stream>>>(HOUT, outp + oofs[j], V, B, Co, tot);
        }
        // pool to next level's input
        if (j < 5) {
            size_t tot = (size_t)(V / 2) * B * Co;
            k_pool2<<<cdiv_u(tot, TPB), TPB, 0, stream>>>(HOUT, XIN, V / 2, B, Co, tot);
        }
    }
}